// GATLayer_48782238548193
// MI455X (gfx1250) — compile-verified
//
#include <hip/hip_runtime.h>
#include <hip/hip_bf16.h>

typedef __attribute__((ext_vector_type(2))) float v2f;
typedef __attribute__((ext_vector_type(8))) float v8f;

#define GAT_N   4096
#define GAT_E   131072
#define GAT_INF 512
#define GAT_H   8
#define GAT_F   32
#define GAT_HF  256        // H*F
#define SLOPE   0.2f
#define TILE_K  128        // K-chunk staged in LDS
#define TILE_N  128        // column panel per block
// LDS: TILE_K x TILE_N fp32, k-pair interleaved = 64 KB

// ---------------------------------------------------------------------------
// Kernel 1: g = h @ W_fc   [4096,512] x [512,256] -> [4096,256]
// fp32 WMMA 16x16x4. Block = 128 threads (4 waves) -> 64x128 output tile.
// Each wave: 16 rows x 128 cols = 8 v8f accumulators; one A frag feeds 8 WMMAs.
// LDS holds W chunk with K-pairs interleaved so each B fragment is ONE
// ds_load_b64 into an even-aligned VGPR pair (no repack movs).
//   element (k, c)  ->  wlds[(k>>1)*(2*TILE_N) + 2*c + (k&1)]
// ---------------------------------------------------------------------------
__global__ __launch_bounds__(128) void gat_gemm_wmma(
    const float* __restrict__ h, const float* __restrict__ Wfc,
    float* __restrict__ g)
{
    __shared__ float wlds[TILE_K * TILE_N];

    const int tid   = threadIdx.x;          // 0..127
    const int wave  = tid >> 5;             // 0..3
    const int lane  = tid & 31;
    const int row0  = blockIdx.x * 64 + wave * 16;
    const int col0  = blockIdx.y * TILE_N;

    const int m     = lane & 15;            // M index (A) / N index (B,D)
    const int koff  = (lane >> 4) << 1;     // lanes 0-15 -> K=0,1 ; 16-31 -> K=2,3
    const int k2off = lane >> 4;            // k-pair offset for B LDS reads

    v8f zero = {0.f,0.f,0.f,0.f,0.f,0.f,0.f,0.f};
    v8f acc[8];
    #pragma unroll
    for (int n = 0; n < 8; ++n) acc[n] = zero;

    for (int kc = 0; kc < GAT_INF; kc += TILE_K) {
        // Stage W_fc[kc:kc+128, col0:col0+128] into LDS, k-pair interleaved.
        // Coalesced global reads: 128 consecutive cols per row.
        #pragma unroll
        for (int i = 0; i < (TILE_K * TILE_N) / 128; ++i) {
            int idx = tid + i * 128;
            int kr  = idx >> 7;              // 0..127
            int cc  = idx & 127;             // 0..127
            wlds[(kr >> 1) * (2 * TILE_N) + 2 * cc + (kr & 1)] =
                Wfc[(size_t)(kc + kr) * GAT_HF + col0 + cc];
        }
        __syncthreads();

        #pragma unroll 4
        for (int k4 = 0; k4 < TILE_K; k4 += 4) {
            // A fragment: 16x4 fp32, 2 contiguous floats per lane (b64 load)
            v2f a = *(const v2f*)(h + (size_t)(row0 + m) * GAT_INF + kc + k4 + koff);
            const int k2 = (k4 >> 1) + k2off;
            #pragma unroll
            for (int n = 0; n < 8; ++n) {
                int c = n * 16 + m;
                v2f b = *(const v2f*)&wlds[k2 * (2 * TILE_N) + 2 * c];
                acc[n] = __builtin_amdgcn_wmma_f32_16x16x4_f32(
                    /*neg_a=*/false, a, /*neg_b=*/false, b,
                    /*c_mod=*/(short)0, acc[n],
                    /*reuse_a=*/false, /*reuse_b=*/false);
            }
        }
        __syncthreads();
    }

    // D layout: lanes 0-15 -> rows M=0..7, lanes 16-31 -> rows M=8..15
    const int rbase = row0 + ((lane >> 4) << 3);
    #pragma unroll
    for (int n = 0; n < 8; ++n) {
        int c = col0 + n * 16 + m;
        #pragma unroll
        for (int r = 0; r < 8; ++r)
            g[(size_t)(rbase + r) * GAT_HF + c] = acc[n][r];
    }
}

// ---------------------------------------------------------------------------
// Kernel 2: el[i,h] = sum_f g[i,h,f]*W_attn[f], er[i,h] = sum_f g[i,h,f]*W_attn[F+f]
// One wave per node; lanes = f dimension; wave32 shuffle reduction.
// ---------------------------------------------------------------------------
__global__ __launch_bounds__(256) void gat_attn_proj(
    const float* __restrict__ g, const float* __restrict__ Wattn,
    float* __restrict__ el, float* __restrict__ er)
{
    int node = (blockIdx.x * blockDim.x + threadIdx.x) >> 5;
    int lane = threadIdx.x & 31;
    if (node >= GAT_N) return;

    float wa = Wattn[lane];
    float wb = Wattn[GAT_F + lane];
    const float* gi = g + (size_t)node * GAT_HF;

    #pragma unroll
    for (int hh = 0; hh < GAT_H; ++hh) {
        float gv = gi[hh * GAT_F + lane];
        float s1 = gv * wa;
        float s2 = gv * wb;
        #pragma unroll
        for (int off = 16; off > 0; off >>= 1) {
            s1 += __shfl_xor(s1, off, 32);
            s2 += __shfl_xor(s2, off, 32);
        }
        if (lane == 0) {
            el[node * GAT_H + hh] = s1;
            er[node * GAT_H + hh] = s2;
        }
    }
}

// ---------------------------------------------------------------------------
// Kernel 3: per-edge scatter. One wave per edge (incl. self-loops).
//   w_h = exp(leaky_relu(el[src,h] + er[dst,h]))
//   num[src,h,:] += w_h * g[dst,h,:]   (32 lanes = F dim)
//   denom[src,h] += w_h
// g/num fit in L2 (4 MB each of 192 MB) -> atomics stay on-chip.
// ---------------------------------------------------------------------------
__global__ __launch_bounds__(256) void gat_edge_scatter(
    const int* __restrict__ ei, const float* __restrict__ mask,
    const float* __restrict__ el, const float* __restrict__ er,
    const float* __restrict__ g, float* __restrict__ num,
    float* __restrict__ denom)
{
    int e    = (blockIdx.x * blockDim.x + threadIdx.x) >> 5;
    int lane = threadIdx.x & 31;
    const int total = GAT_E + GAT_N;
    if (e >= total) return;

    int src, dst;
    if (e < GAT_E) { src = ei[e]; dst = ei[GAT_E + e]; }   // edge_index[0,:],[1,:]
    else           { src = dst = e - GAT_E; }              // appended self-loop

    // adjacency value = train_mask[e]; zero disables edge (diag always on)
    if (mask[e] == 0.0f && src != dst) return;

    const float* gd = g + (size_t)dst * GAT_HF;
    #pragma unroll
    for (int hh = 0; hh < GAT_H; ++hh) {
        float logit = el[src * GAT_H + hh] + er[dst * GAT_H + hh];
        float lr = logit > 0.f ? logit : SLOPE * logit;
        float w  = __expf(lr);
        atomicAdd(&num[(size_t)src * GAT_HF + hh * GAT_F + lane],
                  w * gd[hh * GAT_F + lane]);
        if (lane == 0)
            atomicAdd(&denom[src * GAT_H + hh], w);
    }
}

// ---------------------------------------------------------------------------
// Kernel 4: out[i,h,f] = num[i,h,f] / denom[i,h]
// ---------------------------------------------------------------------------
__global__ __launch_bounds__(256) void gat_normalize(
    float* __restrict__ out, const float* __restrict__ denom)
{
    int t = blockIdx.x * blockDim.x + threadIdx.x;
    if (t >= GAT_N * GAT_HF) return;
    int i  = t >> 8;          // node
    int hh = (t >> 5) & 7;    // head
    out[t] /= denom[i * GAT_H + hh];
}

// ---------------------------------------------------------------------------
extern "C" void kernel_launch(void* const* d_in, const int* in_sizes, int n_in,
                              void* d_out, int out_size, void* d_ws, size_t ws_size,
                              hipStream_t stream) {
    const float* h     = (const float*)d_in[0];   // [4096, 512]
    const int*   ei    = (const int*)  d_in[1];   // [2, 131072]
    const float* mask  = (const float*)d_in[2];   // [E + N]
    const float* Wfc   = (const float*)d_in[3];   // [512, 256]
    const float* Wattn = (const float*)d_in[4];   // [64]
    float*       out   = (float*)d_out;           // [4096, 256]

    // workspace layout (floats): g | el | er | denom
    float* g     = (float*)d_ws;
    float* el    = g  + (size_t)GAT_N * GAT_HF;
    float* er    = el + (size_t)GAT_N * GAT_H;
    float* denom = er + (size_t)GAT_N * GAT_H;

    hipMemsetAsync(out,   0, (size_t)GAT_N * GAT_HF * sizeof(float), stream);
    hipMemsetAsync(denom, 0, (size_t)GAT_N * GAT_H  * sizeof(float), stream);

    dim3 ggrid(GAT_N / 64, GAT_HF / TILE_N);   // 64 x 2 blocks
    gat_gemm_wmma<<<ggrid, 128, 0, stream>>>(h, Wfc, g);

    gat_attn_proj<<<(GAT_N * 32 + 255) / 256, 256, 0, stream>>>(g, Wattn, el, er);

    const int totalE = GAT_E + GAT_N;
    gat_edge_scatter<<<((size_t)totalE * 32 + 255) / 256, 256, 0, stream>>>(
        ei, mask, el, er, g, out, denom);

    gat_normalize<<<(GAT_N * GAT_HF + 255) / 256, 256, 0, stream>>>(out, denom);
}